// Text_GCN_59828894433328
// MI455X (gfx1250) — compile-verified
//
#include <hip/hip_runtime.h>

typedef __attribute__((ext_vector_type(2))) float v2f;
typedef __attribute__((ext_vector_type(8))) float v8f;
typedef __attribute__((ext_vector_type(4))) float f32x4;

#define GCN_IN_CH  256
#define GCN_HID_CH 128
#define GCN_OUT_CH 64

// ---------------------------------------------------------------------------
// GEMM1: h[N,128] = x[N,256] @ W1[256,128]  via V_WMMA_F32_16X16X4_F32
// Block = 256 threads = 8 waves. Block owns 16 rows (M-tile); wave w owns
// output columns [16w, 16w+16). LDS-staged x tile with padded stride.
// ---------------------------------------------------------------------------
#define XS_LD (GCN_IN_CH + 4)

__global__ __launch_bounds__(256) void gcn_gemm1(const float* __restrict__ x,
                                                 const float* __restrict__ W,
                                                 float* __restrict__ h) {
    __shared__ float xs[16 * XS_LD];
    const int m0  = blockIdx.x * 16;
    const int tid = threadIdx.x;

    #pragma unroll
    for (int i = 0; i < 4; ++i) {
        const int flat = tid + i * 256;
        const int row  = flat >> 6;
        const int c4   = (flat & 63) * 4;
        f32x4 v = *(const f32x4*)(x + (size_t)(m0 + row) * GCN_IN_CH + c4);
        *(f32x4*)(xs + row * XS_LD + c4) = v;
    }
    __syncthreads();

    const int lane = tid & 31;
    const int hi   = lane >> 4;            // 0: K=0,1  1: K=2,3 (ISA A layout)
    const int l16  = lane & 15;
    const int col  = (tid >> 5) * 16 + l16;

    v8f c = {};
    #pragma unroll 4
    for (int k = 0; k < GCN_IN_CH; k += 4) {
        const int kk = k + hi * 2;
        v2f a, b;
        a.x = xs[l16 * XS_LD + kk];
        a.y = xs[l16 * XS_LD + kk + 1];
        b.x = W[(size_t)kk * GCN_HID_CH + col];
        b.y = W[(size_t)(kk + 1) * GCN_HID_CH + col];
        c = __builtin_amdgcn_wmma_f32_16x16x4_f32(false, a, false, b,
                                                  (short)0, c, false, false);
    }
    #pragma unroll
    for (int r = 0; r < 8; ++r)
        h[(size_t)(m0 + hi * 8 + r) * GCN_HID_CH + col] = c[r];
}

// ---------------------------------------------------------------------------
// GEMM2: h2[N,64] = relu(agg1)[N,128] @ W2[128,64]. ReLU fused into staging.
// ---------------------------------------------------------------------------
#define HS_LD (GCN_HID_CH + 4)

__global__ __launch_bounds__(128) void gcn_gemm2(const float* __restrict__ hin,
                                                 const float* __restrict__ W,
                                                 float* __restrict__ hout) {
    __shared__ float hs[16 * HS_LD];
    const int m0  = blockIdx.x * 16;
    const int tid = threadIdx.x;

    #pragma unroll
    for (int i = 0; i < 4; ++i) {
        const int flat = tid + i * 128;
        const int row  = flat >> 5;
        const int c4   = (flat & 31) * 4;
        f32x4 v = *(const f32x4*)(hin + (size_t)(m0 + row) * GCN_HID_CH + c4);
        v.x = fmaxf(v.x, 0.0f); v.y = fmaxf(v.y, 0.0f);
        v.z = fmaxf(v.z, 0.0f); v.w = fmaxf(v.w, 0.0f);
        *(f32x4*)(hs + row * HS_LD + c4) = v;
    }
    __syncthreads();

    const int lane = tid & 31;
    const int hi   = lane >> 4;
    const int l16  = lane & 15;
    const int col  = (tid >> 5) * 16 + l16;

    v8f c = {};
    #pragma unroll 4
    for (int k = 0; k < GCN_HID_CH; k += 4) {
        const int kk = k + hi * 2;
        v2f a, b;
        a.x = hs[l16 * HS_LD + kk];
        a.y = hs[l16 * HS_LD + kk + 1];
        b.x = W[(size_t)kk * GCN_OUT_CH + col];
        b.y = W[(size_t)(kk + 1) * GCN_OUT_CH + col];
        c = __builtin_amdgcn_wmma_f32_16x16x4_f32(false, a, false, b,
                                                  (short)0, c, false, false);
    }
    #pragma unroll
    for (int r = 0; r < 8; ++r)
        hout[(size_t)(m0 + hi * 8 + r) * GCN_OUT_CH + col] = c[r];
}

// ---------------------------------------------------------------------------
// CSR build: degree histogram -> exclusive scan -> fill (col, weight).
// ---------------------------------------------------------------------------
__global__ __launch_bounds__(256) void gcn_zero_u32(unsigned* __restrict__ p, int n) {
    const int i = blockIdx.x * 256 + threadIdx.x;
    if (i < n) p[i] = 0u;
}

__global__ __launch_bounds__(256) void gcn_degree(const int* __restrict__ dst,
                                                  unsigned* __restrict__ deg,
                                                  long long E) {
    const long long e = (long long)blockIdx.x * 256 + threadIdx.x;
    if (e < E) atomicAdd(&deg[dst[e]], 1u);
}

// Single-block exclusive scan over deg[0..n): writes rowptr[0..n] and pos copy.
__global__ __launch_bounds__(1024) void gcn_scan(const unsigned* __restrict__ deg,
                                                 unsigned* __restrict__ rowptr,
                                                 unsigned* __restrict__ pos, int n) {
    __shared__ unsigned part[1024];
    const int tid  = threadIdx.x;
    const int per  = (n + 1023) / 1024;
    const int base = tid * per;

    unsigned sum = 0;
    for (int i = 0; i < per; ++i) {
        const int idx = base + i;
        if (idx < n) sum += deg[idx];
    }
    part[tid] = sum;
    __syncthreads();
    for (int off = 1; off < 1024; off <<= 1) {
        const unsigned v = (tid >= off) ? part[tid - off] : 0u;
        __syncthreads();
        part[tid] += v;
        __syncthreads();
    }
    unsigned run = (tid == 0) ? 0u : part[tid - 1];
    for (int i = 0; i < per; ++i) {
        const int idx = base + i;
        if (idx < n) {
            rowptr[idx] = run;
            pos[idx]    = run;
            run += deg[idx];
        }
    }
    if (tid == 1023) rowptr[n] = part[1023];
}

__global__ __launch_bounds__(256) void gcn_fill(const int* __restrict__ src,
                                                const int* __restrict__ dst,
                                                const float* __restrict__ ew,
                                                unsigned* __restrict__ pos,
                                                int* __restrict__ col,
                                                float* __restrict__ wv,
                                                long long E) {
    const long long e = (long long)blockIdx.x * 256 + threadIdx.x;
    if (e >= E) return;
    const unsigned idx = atomicAdd(&pos[dst[e]], 1u);
    col[idx] = src[e];
    wv[idx]  = ew[e];
}

// ---------------------------------------------------------------------------
// Gather aggregation: one wave per node. out[node] = bias + sum_j w_j*h[col_j].
// Edge (src, w) pairs loaded coalesced in chunks of 32, broadcast via __shfl
// (ds_bpermute on wave32). No float atomics; accumulators stay in registers.
// ---------------------------------------------------------------------------
template <int CH>
__global__ __launch_bounds__(256) void gcn_gather(const float* __restrict__ h,
                                                  const int* __restrict__ col,
                                                  const float* __restrict__ wv,
                                                  const unsigned* __restrict__ rowptr,
                                                  const float* __restrict__ bias,
                                                  float* __restrict__ out, int n) {
    constexpr int FPL = CH / 32;   // floats per lane: 4 (CH=128) or 2 (CH=64)
    const int lane = threadIdx.x & 31;
    const int node = blockIdx.x * 8 + (threadIdx.x >> 5);
    if (node >= n) return;

    const unsigned beg = rowptr[node];
    const unsigned end = rowptr[node + 1];
    const int c0 = lane * FPL;

    float acc[FPL];
    #pragma unroll
    for (int k = 0; k < FPL; ++k) acc[k] = 0.0f;

    for (unsigned j = beg; j < end; j += 32u) {
        const unsigned jj = j + (unsigned)lane;
        int s = 0; float w = 0.0f;
        if (jj < end) { s = col[jj]; w = wv[jj]; }
        const unsigned rem = end - j;
        const int cnt = rem < 32u ? (int)rem : 32;
        for (int i = 0; i < cnt; ++i) {
            const int   si = __shfl(s, i, 32);
            const float wi = __shfl(w, i, 32);
            const float* hp = h + (size_t)si * CH + c0;
            if constexpr (FPL == 4) {
                const f32x4 v = *(const f32x4*)hp;
                acc[0] += wi * v.x; acc[1] += wi * v.y;
                acc[2] += wi * v.z; acc[3] += wi * v.w;
            } else {
                const v2f v = *(const v2f*)hp;
                acc[0] += wi * v.x; acc[1] += wi * v.y;
            }
        }
    }
    float* op = out + (size_t)node * CH + c0;
    #pragma unroll
    for (int k = 0; k < FPL; ++k) op[k] = bias[c0 + k] + acc[k];
}

// ---------------------------------------------------------------------------
extern "C" void kernel_launch(void* const* d_in, const int* in_sizes, int n_in,
                              void* d_out, int out_size, void* d_ws, size_t ws_size,
                              hipStream_t stream) {
    const float* x  = (const float*)d_in[0];   // [N, 256]
    const int*   ei = (const int*)d_in[1];     // [2, E]: src row then dst row
    const float* ew = (const float*)d_in[2];   // [E]
    const float* W1 = (const float*)d_in[3];   // [256, 128]
    const float* b1 = (const float*)d_in[4];   // [128]
    const float* W2 = (const float*)d_in[5];   // [128, 64]
    const float* b2 = (const float*)d_in[6];   // [64]
    float* out = (float*)d_out;                // [N, 64]

    const int       N = in_sizes[0] / GCN_IN_CH;  // 50000, divisible by 16
    const long long E = in_sizes[2];              // 1,600,000

    const int* src = ei;
    const int* dst = ei + E;

    // Workspace layout
    float*    h1     = (float*)d_ws;                           // N*128
    float*    agg1   = h1 + (size_t)N * GCN_HID_CH;            // N*128
    float*    h2     = agg1 + (size_t)N * GCN_HID_CH;          // N*64
    unsigned* deg    = (unsigned*)(h2 + (size_t)N * GCN_OUT_CH); // N
    unsigned* rowptr = deg + N;                                // N+1
    unsigned* pos    = rowptr + N + 1;                         // N
    int*      col    = (int*)(pos + N);                        // E
    float*    wv     = (float*)(col + E);                      // E

    const unsigned eb = (unsigned)((E + 255) / 256);

    // CSR build (by destination)
    gcn_zero_u32<<<(N + 255) / 256, 256, 0, stream>>>(deg, N);
    gcn_degree<<<eb, 256, 0, stream>>>(dst, deg, E);
    gcn_scan<<<1, 1024, 0, stream>>>(deg, rowptr, pos, N);
    gcn_fill<<<eb, 256, 0, stream>>>(src, dst, ew, pos, col, wv, E);

    // Layer 1: h1 = x @ W1 ; agg1 = b1 + segment_sum(w * h1[src])
    gcn_gemm1<<<N / 16, 256, 0, stream>>>(x, W1, h1);
    gcn_gather<GCN_HID_CH><<<(N + 7) / 8, 256, 0, stream>>>(
        h1, col, wv, rowptr, b1, agg1, N);

    // Layer 2: h2 = relu(agg1) @ W2 ; out = b2 + segment_sum(w * h2[src])
    gcn_gemm2<<<N / 16, 128, 0, stream>>>(agg1, W2, h2);
    gcn_gather<GCN_OUT_CH><<<(N + 7) / 8, 256, 0, stream>>>(
        h2, col, wv, rowptr, b2, out, N);
}